// UnivNetLVCBlock_29540785062324
// MI455X (gfx1250) — compile-verified
//
#include <hip/hip_runtime.h>
#include <hip/hip_bf16.h>

typedef float v2f __attribute__((ext_vector_type(2)));
typedef float v8f __attribute__((ext_vector_type(8)));

__device__ __forceinline__ float lrelu(float x) { return x >= 0.f ? x : 0.2f * x; }

// ---------------- 1. conv_transpose1d: x(4,32,8192) -> H0(4,32,65536) ----------------
// y[b,o,t] = b[o] + sum_c w[c,o,15-k0]*x[b,c,m0] + w[c,o,7-k0]*x[b,c,m0+1]
// with k0 = (11 - t) mod 8, m0 = (t - 11 + k0)/8
__global__ __launch_bounds__(256) void k_convt(const float* __restrict__ x,
                                               const float* __restrict__ w,
                                               const float* __restrict__ bias,
                                               float* __restrict__ y) {
    int idx = blockIdx.x * 256 + threadIdx.x;      // (b*32+o)*65536 + t
    int t  = idx & 65535;
    int bo = idx >> 16;
    int o  = bo & 31, b = bo >> 5;
    int k0 = ((11 - t) % 8 + 8) % 8;
    int m0 = (t - 11 + k0) >> 3;                   // exact (arithmetic shift ok)
    float acc = bias[o];
    const float* xb = x + (size_t)b * 32 * 8192;
    if (m0 >= 0 && m0 < 8192) {
        int wi = 15 - k0;
        for (int c = 0; c < 32; ++c)
            acc += w[(c * 32 + o) * 16 + wi] * xb[c * 8192 + m0];
    }
    int m1 = m0 + 1;
    if (m1 >= 0 && m1 < 8192) {
        int wi = 7 - k0;
        for (int c = 0; c < 32; ++c)
            acc += w[(c * 32 + o) * 16 + wi] * xb[c * 8192 + m1];
    }
    y[idx] = acc;
}

// ---------------- 2. kp_in conv: spec(4,100,256) -> (4,64,256), K=5 pad 2 ----------------
__global__ __launch_bounds__(256) void k_kp_in(const float* __restrict__ spec,
                                               const float* __restrict__ w,
                                               const float* __restrict__ bias,
                                               float* __restrict__ out) {
    int idx = blockIdx.x * 256 + threadIdx.x;      // (b*64+oc)*256 + l
    int l = idx & 255;
    int oc = (idx >> 8) & 63;
    int b = idx >> 14;
    float acc = bias[oc];
    const float* sb = spec + (size_t)b * 100 * 256;
    const float* wr = w + oc * 100 * 5;
    for (int ic = 0; ic < 100; ++ic)
        for (int k = 0; k < 5; ++k) {
            int ll = l + k - 2;
            if (ll >= 0 && ll < 256) acc += wr[ic * 5 + k] * sb[ic * 256 + ll];
        }
    out[idx] = acc;
}

// ---------------- 3. 64ch conv3 + lrelu (+ optional residual) ----------------
__global__ __launch_bounds__(256) void k_conv3_64(const float* __restrict__ in,
                                                  const float* __restrict__ w,
                                                  const float* __restrict__ bias,
                                                  const float* __restrict__ res,
                                                  float* __restrict__ out) {
    int idx = blockIdx.x * 256 + threadIdx.x;      // (b*64+oc)*256 + l
    int l = idx & 255;
    int oc = (idx >> 8) & 63;
    int b = idx >> 14;
    float acc = bias[oc];
    const float* ib = in + (size_t)b * 64 * 256;
    const float* wr = w + oc * 64 * 3;
    for (int ic = 0; ic < 64; ++ic) {
        float w0 = wr[ic * 3], w1 = wr[ic * 3 + 1], w2 = wr[ic * 3 + 2];
        if (l > 0)   acc += w0 * ib[ic * 256 + l - 1];
        acc += w1 * ib[ic * 256 + l];
        if (l < 255) acc += w2 * ib[ic * 256 + l + 1];
    }
    acc = lrelu(acc);
    if (res) acc += res[idx];
    out[idx] = acc;
}

// ---------------- 4. kern conv as WMMA f32 GEMM, transposed store ----------------
// A = kern_w row-major [24576 x 192]; B[j][l] = h[b][j/3][l + j%3 - 1]
// C tile (128 x 64) is transposed through LDS and stored to KERNT[b][l][kc] coalesced.
#define KBS 65
#define CTS 65
__global__ __launch_bounds__(256) void k_kern_wmma(const float* __restrict__ A,
                                                   const float* __restrict__ kb,
                                                   const float* __restrict__ hb,
                                                   float* __restrict__ outT) {
    __shared__ float Bl[192 * KBS];                 // 49920 B; reused as 128x64 C tile (8320 floats)
    float* Ct = Bl;
    int b = blockIdx.z;
    int l0 = blockIdx.y * 64;
    int Mblk = blockIdx.x * 128;
    const float* h = hb + (size_t)b * 64 * 256;
    for (int idx = threadIdx.x; idx < 192 * 64; idx += 256) {   // 48 iters exact
        int j = idx >> 6, n = idx & 63;
        int hh = j / 3, k = j - hh * 3;
        int ll = l0 + n + k - 1;
        Bl[j * KBS + n] = (ll >= 0 && ll < 256) ? h[hh * 256 + ll] : 0.f;
    }
    __syncthreads();
    int lane = threadIdx.x & 31;
    int wave = threadIdx.x >> 5;
    int lhalf = lane >> 4, l16 = lane & 15;
    int Mbase = Mblk + wave * 16;
    v8f acc[4] = {};
    const float* arow = A + (size_t)(Mbase + l16) * 192 + lhalf * 2;
    for (int kk = 0; kk < 192; kk += 4) {
        v2f a = *(const v2f*)(arow + kk);                       // A[M][kk + 2*lhalf .. +1]
        int jb = kk + lhalf * 2;
#pragma unroll
        for (int nt = 0; nt < 4; ++nt) {
            v2f f;
            f.x = Bl[jb * KBS + nt * 16 + l16];                 // B[K=jb  ][n]
            f.y = Bl[(jb + 1) * KBS + nt * 16 + l16];           // B[K=jb+1][n]
            acc[nt] = __builtin_amdgcn_wmma_f32_16x16x4_f32(
                false, a, false, f, (short)0, acc[nt], false, false);
        }
    }
    __syncthreads();                                            // Bl reads done, reuse as Ct
#pragma unroll
    for (int nt = 0; nt < 4; ++nt) {
#pragma unroll
        for (int i = 0; i < 8; ++i) {
            int rloc = wave * 16 + i + lhalf * 8;               // 0..127
            Ct[rloc * CTS + nt * 16 + l16] = acc[nt][i] + kb[Mblk + rloc];
        }
    }
    __syncthreads();
    float* ob = outT + (size_t)b * 256 * 24576;
    for (int idx = threadIdx.x; idx < 64 * 128; idx += 256) {   // 32 iters exact
        int n = idx >> 7, m = idx & 127;                        // n: local l, m: local kc
        ob[(size_t)(l0 + n) * 24576 + Mblk + m] = Ct[m * CTS + n];
    }
}

// ---------------- 5. bias conv: h(4,64,256) -> BIASB(4,256,256) ----------------
__global__ __launch_bounds__(256) void k_bias_conv(const float* __restrict__ hb,
                                                   const float* __restrict__ w,
                                                   const float* __restrict__ bias,
                                                   float* __restrict__ out) {
    int idx = blockIdx.x * 256 + threadIdx.x;      // (b*256+bc)*256 + l
    int l = idx & 255;
    int bc = (idx >> 8) & 255;
    int b = idx >> 16;
    float acc = bias[bc];
    const float* ib = hb + (size_t)b * 64 * 256;
    const float* wr = w + bc * 192;
    for (int ic = 0; ic < 64; ++ic) {
        if (l > 0)   acc += wr[ic * 3 + 0] * ib[ic * 256 + l - 1];
        acc += wr[ic * 3 + 1] * ib[ic * 256 + l];
        if (l < 255) acc += wr[ic * 3 + 2] * ib[ic * 256 + l + 1];
    }
    out[idx] = acc;
}

// ---------------- 6. lvc dilated conv: Y = lrelu(conv(lrelu(H), w, dil)) ----------------
__global__ __launch_bounds__(256) void k_lvc_conv(const float* __restrict__ H,
                                                  const float* __restrict__ w,
                                                  const float* __restrict__ bias,
                                                  float* __restrict__ y, int dil) {
    int idx = blockIdx.x * 256 + threadIdx.x;      // (b*32+c)*65536 + t
    int t = idx & 65535;
    int c = (idx >> 16) & 31;
    int b = idx >> 21;
    float acc = bias[c];
    const float* hb = H + (size_t)b * 32 * 65536;
    const float* wr = w + c * 32 * 3;
    int tm = t - dil, tp = t + dil;
    for (int ic = 0; ic < 32; ++ic) {
        const float* hc = hb + (size_t)ic * 65536;
        if (tm >= 0)     acc += wr[ic * 3 + 0] * lrelu(hc[tm]);
        acc += wr[ic * 3 + 1] * lrelu(hc[t]);
        if (tp < 65536)  acc += wr[ic * 3 + 2] * lrelu(hc[tp]);
    }
    y[idx] = lrelu(acc);
}

// ---------------- 7. LVC WMMA GEMM + bias + sigmoid*tanh gate + residual ----------------
// Per (b,l): C[o=64][s=256] = sum_{j=0..95} A[o][j]*B[j][s]
// A[o][c*3+k] = KERNT[b][l][dil*6144 + c*192 + o*3 + k]  (contiguous 6144-float slice)
// B[c*3+k][s] = Y[b][c][l*256 + s + k - 1]
// Hout[ch] = sigmoid(C[ch]+bias)*tanh(C[ch+32]+bias) + Hin[ch]
#define AS 100
#define XS 264
__global__ __launch_bounds__(256) void k_lvc_wmma(const float* __restrict__ Y,
                                                  const float* __restrict__ KERNT,
                                                  const float* __restrict__ BIASb,
                                                  const float* __restrict__ Hin,
                                                  float* __restrict__ Hout, int dil) {
    __shared__ float Al[64 * AS];
    __shared__ float Xl[32 * XS];
    int l = blockIdx.x;
    int b = blockIdx.y;
    const float* hin = Hin + (size_t)b * 32 * 65536;
    // prefetch residual rows (consumed in epilogue) while we stage LDS + run the GEMM
    {
        int pr = threadIdx.x >> 3;                  // 0..31 channel
        int po = (threadIdx.x & 7) * 32;            // 0..224 offset
        __builtin_prefetch(hin + (size_t)pr * 65536 + l * 256 + po, 0, 1);
    }
    // A slice: fully coalesced contiguous load, scattered into [o][j] LDS layout
    const float* kbase = KERNT + ((size_t)b * 256 + l) * 24576 + (size_t)dil * 6144;
    for (int idx = threadIdx.x; idx < 6144; idx += 256) {       // 24 iters exact
        int c = idx / 192;
        int r = idx - c * 192;
        int o = r / 3, k = r - o * 3;
        Al[o * AS + c * 3 + k] = kbase[idx];
    }
    const float* yb = Y + (size_t)b * 32 * 65536;
    int t0 = l * 256 - 1;
    for (int idx = threadIdx.x; idx < 32 * 258; idx += 256) {
        int c = idx / 258, s = idx - c * 258;
        int t = t0 + s;
        Xl[c * XS + s] = (t >= 0 && t < 65536) ? yb[(size_t)c * 65536 + t] : 0.f;
    }
    __syncthreads();
    int lane = threadIdx.x & 31;
    int wave = threadIdx.x >> 5;
    int lhalf = lane >> 4, l16 = lane & 15;
    int sb = wave * 32;                                         // each wave: 32 cols, full 64 rows
    v8f acc[4][2] = {};
    for (int kk = 0; kk < 96; kk += 4) {
        int j0 = kk + lhalf * 2;
        int c0 = j0 / 3, k0 = j0 - c0 * 3;
        int j1 = j0 + 1;
        int c1 = j1 / 3, k1 = j1 - c1 * 3;
        v2f bf[2];
#pragma unroll
        for (int nt = 0; nt < 2; ++nt) {
            int s = sb + nt * 16 + l16;
            bf[nt].x = Xl[c0 * XS + s + k0];
            bf[nt].y = Xl[c1 * XS + s + k1];
        }
#pragma unroll
        for (int mt = 0; mt < 4; ++mt) {
            v2f a = *(const v2f*)&Al[(mt * 16 + l16) * AS + kk + lhalf * 2];
#pragma unroll
            for (int nt = 0; nt < 2; ++nt)
                acc[mt][nt] = __builtin_amdgcn_wmma_f32_16x16x4_f32(
                    false, a, false, bf[nt], (short)0, acc[mt][nt], false, false);
        }
    }
    const float* biasrow = BIASb + ((size_t)b * 4 + dil) * 64 * 256 + l;
    float* hout = Hout + (size_t)b * 32 * 65536;
#pragma unroll
    for (int mt = 0; mt < 2; ++mt) {
#pragma unroll
        for (int nt = 0; nt < 2; ++nt) {
            int col = sb + nt * 16 + l16;
            int tg = l * 256 + col;
#pragma unroll
            for (int i = 0; i < 8; ++i) {
                int row = mt * 16 + i + lhalf * 8;              // sigmoid channel (0..31)
                float s_in = acc[mt][nt][i] + biasrow[(size_t)row * 256];
                float t_in = acc[mt + 2][nt][i] + biasrow[(size_t)(row + 32) * 256];
                float sg = 1.f / (1.f + __expf(-s_in));
                float th = tanhf(t_in);
                hout[(size_t)row * 65536 + tg] = sg * th + hin[(size_t)row * 65536 + tg];
            }
        }
    }
}

// ---------------- 8. final copy H0 -> out ----------------
__global__ __launch_bounds__(256) void k_copy4(const float4* __restrict__ src,
                                               float4* __restrict__ dst) {
    int i = blockIdx.x * 256 + threadIdx.x;
    dst[i] = src[i];
}

extern "C" void kernel_launch(void* const* d_in, const int* in_sizes, int n_in,
                              void* d_out, int out_size, void* d_ws, size_t ws_size,
                              hipStream_t stream) {
    (void)in_sizes; (void)n_in; (void)out_size; (void)ws_size;
    const float* hs      = (const float*)d_in[0];
    const float* spec    = (const float*)d_in[1];
    const float* convt_w = (const float*)d_in[2];
    const float* convt_b = (const float*)d_in[3];
    const float* kp_in_w = (const float*)d_in[4];
    const float* kp_in_b = (const float*)d_in[5];
    const float* rb_w1   = (const float*)d_in[6];
    const float* rb_b1   = (const float*)d_in[7];
    const float* rb_w2   = (const float*)d_in[8];
    const float* rb_b2   = (const float*)d_in[9];
    const float* kern_w  = (const float*)d_in[10];
    const float* kern_b  = (const float*)d_in[11];
    const float* bias_w  = (const float*)d_in[12];
    const float* bias_b  = (const float*)d_in[13];
    const float* lvc_w   = (const float*)d_in[14];
    const float* lvc_b   = (const float*)d_in[15];
    float* out = (float*)d_out;

    // workspace layout (floats): H0 | KERNT | BIASB | HA | HB   (~130 MB total)
    float* H0    = (float*)d_ws;
    float* KERNT = H0 + 8388608;        // 4*256*24576  (layout [b][l][kc])
    float* BIASB = KERNT + 25165824;    // 4*256*256
    float* HA    = BIASB + 262144;      // 4*64*256
    float* HB    = HA + 65536;          // 4*64*256
    float* Ybuf  = out;                 // reuse d_out as Y scratch; overwritten at end

    // conv_transpose -> H0
    k_convt<<<32768, 256, 0, stream>>>(hs, convt_w, convt_b, H0);

    // kernel predictor front-end
    k_kp_in<<<256, 256, 0, stream>>>(spec, kp_in_w, kp_in_b, HA);
    for (int i = 0; i < 3; ++i) {
        k_conv3_64<<<256, 256, 0, stream>>>(HA, rb_w1 + i * 64 * 64 * 3, rb_b1 + i * 64, nullptr, HB);
        k_conv3_64<<<256, 256, 0, stream>>>(HB, rb_w2 + i * 64 * 64 * 3, rb_b2 + i * 64, HA, HA);
    }

    // big GEMM (WMMA): kernels tensor, stored transposed [b][l][kc]
    dim3 kgrid(192, 4, 4);
    k_kern_wmma<<<kgrid, 256, 0, stream>>>(kern_w, kern_b, HA, KERNT);
    k_bias_conv<<<1024, 256, 0, stream>>>(HA, bias_w, bias_b, BIASB);

    // main LVC loop
    const int dils[4] = {1, 3, 9, 27};
    for (int i = 0; i < 4; ++i) {
        k_lvc_conv<<<32768, 256, 0, stream>>>(H0, lvc_w + i * 32 * 32 * 3, lvc_b + i * 32, Ybuf, dils[i]);
        dim3 lg(256, 4);
        k_lvc_wmma<<<lg, 256, 0, stream>>>(Ybuf, KERNT, BIASB, H0, H0, i);
    }

    k_copy4<<<8192, 256, 0, stream>>>((const float4*)H0, (float4*)out);
}